// SoftDTW_60309930770596
// MI455X (gfx1250) — compile-verified
//
#include <hip/hip_runtime.h>
#include <math.h>

typedef __attribute__((ext_vector_type(2))) float v2f;
typedef __attribute__((ext_vector_type(8))) float v8f;

constexpr int B = 64, N = 512, M = 512, K = 64;
constexpr int NDIAG = N + M - 1; // 1023
constexpr float F_INF = __builtin_inff();

// Diagonal-packed layout helpers: diag p has cells i in [imin(p), imax(p)],
// stored densely. Total = N*M = 262144 per batch.
__device__ __forceinline__ int diag_imin(int p) { return (p < M) ? 0 : p - (M - 1); }
__device__ __forceinline__ int diag_prefix(int p) {
    return (p < M) ? ((p * (p + 1)) >> 1)
                   : (N * M - (((NDIAG - p) * (NDIAG - p + 1)) >> 1));
}

// ---------------- Kernel 1: squared row norms of x and y ----------------
__global__ void __launch_bounds__(256)
sqnorm_kernel(const float* __restrict__ x, const float* __restrict__ y,
              float* __restrict__ x2, float* __restrict__ y2) {
    int t = blockIdx.x * 256 + threadIdx.x;          // 0 .. B*(N+M)-1
    const float* src; float* dst; int row;
    if (t < B * N) { src = x; dst = x2; row = t; }
    else           { src = y; dst = y2; row = t - B * N; }
    const float4* p = (const float4*)(src + (size_t)row * K);
    float s = 0.0f;
    #pragma unroll
    for (int q = 0; q < K / 4; ++q) {
        float4 v = p[q];
        s += v.x * v.x + v.y * v.y + v.z * v.z + v.w * v.w;
    }
    dst[row] = s;
}

// ---------------- Kernel 2: D = x2 + y2 - 2*x.y^T via WMMA f32 16x16x4,
//                  written in diagonal-packed layout -----------------------
__global__ void __launch_bounds__(256)
pairdist_wmma_kernel(const float* __restrict__ x, const float* __restrict__ y,
                     const float* __restrict__ x2, const float* __restrict__ y2,
                     float* __restrict__ Dd) {
    const int b      = blockIdx.y;
    const int iBlock = (blockIdx.x >> 3) * 64;
    const int jBlock = (blockIdx.x & 7) * 64;
    const int wave   = threadIdx.x >> 5;
    const int lane   = threadIdx.x & 31;
    const int r      = lane & 15;       // row/col within 16
    const int kh     = lane >> 4;       // lane-half selects K group

    const int ti  = wave >> 1;          // 0..3 : 16-row strip
    const int tj0 = (wave & 1) * 2;     // two adjacent 16-col strips

    const float* xb = x + (size_t)b * N * K;
    const float* yb = y + (size_t)b * M * K;

    const int i0 = iBlock + ti * 16;
    const int j0 = jBlock + tj0 * 16 + r;
    const int j1 = j0 + 16;

    const float* arow  = xb + (size_t)(i0 + r) * K;
    const float* bcol0 = yb + (size_t)j0 * K;
    const float* bcol1 = yb + (size_t)j1 * K;

    v8f acc0 = {}; v8f acc1 = {};
    #pragma unroll
    for (int k0 = 0; k0 < K; k0 += 4) {
        // A (16x4): lane-half kh holds K = k0+2kh, k0+2kh+1
        v2f a = *(const v2f*)(arow + k0 + 2 * kh);
        // B (4x16): lane-half kh holds K = k0+kh (v0) and k0+2+kh (v1)
        v2f b0; b0.x = bcol0[k0 + kh]; b0.y = bcol0[k0 + 2 + kh];
        v2f b1; b1.x = bcol1[k0 + kh]; b1.y = bcol1[k0 + 2 + kh];
        acc0 = __builtin_amdgcn_wmma_f32_16x16x4_f32(false, a, false, b0,
                                                     (short)0, acc0, false, false);
        acc1 = __builtin_amdgcn_wmma_f32_16x16x4_f32(false, a, false, b1,
                                                     (short)0, acc1, false, false);
    }

    // Epilogue: D[i][j] = x2[i] + y2[j] - 2*dot, scattered into diag layout.
    const float y2a = y2[b * M + j0];
    const float y2b = y2[b * M + j1];
    float* Db = Dd + (size_t)b * N * M;
    #pragma unroll
    for (int e = 0; e < 8; ++e) {
        int i = i0 + 8 * kh + e;                // C/D layout: M = e + 8*(lane>=16)
        float xv = x2[b * N + i];
        int p0 = i + j0;
        Db[diag_prefix(p0) + (i - diag_imin(p0))] = xv + y2a - 2.0f * acc0[e];
        int p1 = i + j1;
        Db[diag_prefix(p1) + (i - diag_imin(p1))] = xv + y2b - 2.0f * acc1[e];
    }
}

// ---------------- Kernel 3: soft-DTW wavefront DP (gamma = 1) --------------
__global__ void __launch_bounds__(512)
softdtw_dp_kernel(const float* __restrict__ Dd, float* __restrict__ out) {
    const int b = blockIdx.x;
    const int i = threadIdx.x;                  // DP row, 0..N-1
    __shared__ float ring[3][N];
    const float* Db = Dd + (size_t)b * N * M;

    ring[0][i] = F_INF;
    ring[1][i] = F_INF;
    __syncthreads();

    int cur = 2;                                // buffer receiving diag p
    for (int p = 0; p < NDIAG; ++p) {
        const int prev1 = (cur == 0) ? 2 : cur - 1;   // diag p-1
        const int prev2 = (cur == 2) ? 0 : cur + 1;   // diag p-2
        const int j = p - i;
        const bool valid = (j >= 0) && (j < M);

        float d1i = ring[prev1][i];
        float d1m = (i >= 1) ? ring[prev1][i - 1] : F_INF;
        float d2m = (i >= 1) ? ring[prev2][i - 1] : F_INF;

        float r0 = (i >= 1 && j >= 1) ? d2m : ((i == 0 && j == 0) ? 0.0f : F_INF);
        float r1 = (i >= 1) ? d1m : F_INF;
        float r2 = (j >= 1) ? d1i : F_INF;

        float z0 = -r0, z1 = -r1, z2 = -r2;
        float zmax = fmaxf(z0, fmaxf(z1, z2));
        float zs = (zmax > -F_INF) ? zmax : 0.0f;
        float s  = __expf(z0 - zs) + __expf(z1 - zs) + __expf(z2 - zs);
        float softmin = (s > 0.0f) ? -(__logf(s) + zs) : F_INF;

        float nv = F_INF;
        if (valid) {
            nv = Db[diag_prefix(p) + (i - diag_imin(p))] + softmin;
        }
        // Prefetch next diagonal's coalesced slab (global_prefetch_b8).
        int pn = p + 1;
        if (pn < NDIAG && i <= pn && (pn - i) < M) {
            __builtin_prefetch((const void*)(Db + diag_prefix(pn) + (i - diag_imin(pn))), 0, 0);
        }
        ring[cur][i] = nv;
        if (p == NDIAG - 1 && i == N - 1) out[b] = nv;
        __syncthreads();
        cur = prev2;                            // rotate: cur advances (mod 3)
    }
}

// ---------------- Launcher --------------------------------------------------
extern "C" void kernel_launch(void* const* d_in, const int* in_sizes, int n_in,
                              void* d_out, int out_size, void* d_ws, size_t ws_size,
                              hipStream_t stream) {
    const float* x = (const float*)d_in[0];
    const float* y = (const float*)d_in[1];
    float* out = (float*)d_out;

    float* x2 = (float*)d_ws;               // B*N floats
    float* y2 = x2 + B * N;                 // B*M floats
    float* Dd = y2 + B * M;                 // B*N*M floats (diag-packed)

    sqnorm_kernel<<<(B * (N + M)) / 256, 256, 0, stream>>>(x, y, x2, y2);

    dim3 g2((N / 64) * (M / 64), B);        // (64, 64)
    pairdist_wmma_kernel<<<g2, 256, 0, stream>>>(x, y, x2, y2, Dd);

    softdtw_dp_kernel<<<B, N, 0, stream>>>(Dd, out);
}